// deeds_graph_3934190044186
// MI455X (gfx1250) — compile-verified
//
#include <hip/hip_runtime.h>

typedef __attribute__((ext_vector_type(2))) float v2f;
typedef __attribute__((ext_vector_type(8))) float v8f;

#define NVOX (13*13*13)            // 2197
#define NTILE ((NVOX + 15) / 16)   // 138 WMMA tiles of 16 voxels
#define ASZ 5491                   // bufA: max(19*17*17, 17*17*13)
#define BSZ 6137                   // bufB: max(19*19*17, 17*17*17, 17*13*13)

// Use CDNA5 async global->LDS copies (ASYNCcnt) for staging the cost volume.
#define USE_ASYNC_LDS 1

// ---------------------------------------------------------------------------
// Async global->LDS helpers (gfx1250, ISA ch10: GLOBAL_LOAD_ASYNC_TO_LDS_B32)
// ---------------------------------------------------------------------------
__device__ __forceinline__ void async_ld_b32(void* ldsPtr, const float* gptr) {
#if USE_ASYNC_LDS
    unsigned ldsOff = (unsigned)(unsigned long long)ldsPtr;  // low 32 bits of generic LDS ptr = LDS offset
    asm volatile("global_load_async_to_lds_b32 %0, %1, off"
                 :: "v"(ldsOff), "v"(gptr) : "memory");
#else
    *(float*)ldsPtr = *gptr;
#endif
}
__device__ __forceinline__ void async_wait() {
#if USE_ASYNC_LDS
    asm volatile("s_wait_asynccnt 0x0" ::: "memory");
#endif
}

// ---------------------------------------------------------------------------
// K0: w = exp(-dist/200), norm = sum_k w
// ---------------------------------------------------------------------------
__global__ void __launch_bounds__(256) wnorm_kernel(const float* __restrict__ dist,
                                                    float* __restrict__ w,
                                                    float* __restrict__ nrm, int N) {
    int n = blockIdx.x * blockDim.x + threadIdx.x;
    if (n >= N) return;
    float s = 0.f;
#pragma unroll
    for (int k = 0; k < 8; ++k) {
        float ww = __expf(-dist[n * 8 + k] * (1.0f / 200.0f));
        w[n * 8 + k] = ww;
        s += ww;
    }
    nrm[n] = s;
}

// ---------------------------------------------------------------------------
// Separable smooth from 13^3 LDS volume cv (affine a1c + a0c*x applied at load):
//   edge-pad 3 (fused) -> maxpool3 x,y,z -> triangular [1,2,3,2,1]/9 x,y,z
// Each thread owns whole 1-D lines per pass: 1 LDS load per input element.
// Force-inlined so each kernel gets a constant-specialized copy (a0c/a1c).
// ---------------------------------------------------------------------------
__device__ __forceinline__ void smooth_lds(const float* __restrict__ cv,
                                           float* __restrict__ bufA,
                                           float* __restrict__ bufB,
                                           float* __restrict__ outg, int tid,
                                           float a0c, float a1c) {
    const float f0 = 1.f / 9.f, f1 = 2.f / 9.f, f2 = 3.f / 9.f;
    // P1: fused edge-pad + affine + max-x: 361 rows -> bufB (19,19,17)
    for (int r = tid; r < 19 * 19; r += 256) {
        int z = r / 19, y = r % 19;
        int cz = min(max(z - 3, 0), 12), cy = min(max(y - 3, 0), 12);
        const float* row = cv + cz * 169 + cy * 13;
        float rr[13];
#pragma unroll
        for (int j = 0; j < 13; ++j) rr[j] = a1c + a0c * row[j];
        float v[19];
#pragma unroll
        for (int j = 0; j < 19; ++j) {
            int cx = j - 3; cx = cx < 0 ? 0 : (cx > 12 ? 12 : cx);
            v[j] = rr[cx];
        }
        float* o = bufB + r * 17;
#pragma unroll
        for (int j = 0; j < 17; ++j) o[j] = fmaxf(fmaxf(v[j], v[j + 1]), v[j + 2]);
    }
    __syncthreads();
    // P2: max-y: bufB (19,19,17) -> bufA (19,17,17): 323 columns
    for (int c = tid; c < 19 * 17; c += 256) {
        int z = c / 17, x = c % 17;
        const float* in = bufB + z * 323 + x;
        float* o = bufA + z * 289 + x;
        float v[19];
#pragma unroll
        for (int j = 0; j < 19; ++j) v[j] = in[j * 17];
#pragma unroll
        for (int j = 0; j < 17; ++j) o[j * 17] = fmaxf(fmaxf(v[j], v[j + 1]), v[j + 2]);
    }
    __syncthreads();
    // P3: max-z: bufA (19,17,17) -> bufB (17,17,17): 289 columns
    for (int c = tid; c < 17 * 17; c += 256) {
        const float* in = bufA + c;
        float* o = bufB + c;
        float v[19];
#pragma unroll
        for (int j = 0; j < 19; ++j) v[j] = in[j * 289];
#pragma unroll
        for (int j = 0; j < 17; ++j) o[j * 289] = fmaxf(fmaxf(v[j], v[j + 1]), v[j + 2]);
    }
    __syncthreads();
    // P4: tri-x: bufB (17,17,17) -> bufA (17,17,13): 289 rows
    for (int r = tid; r < 17 * 17; r += 256) {
        const float* in = bufB + r * 17;
        float* o = bufA + r * 13;
        float v[17];
#pragma unroll
        for (int j = 0; j < 17; ++j) v[j] = in[j];
#pragma unroll
        for (int j = 0; j < 13; ++j)
            o[j] = f0 * (v[j] + v[j + 4]) + f1 * (v[j + 1] + v[j + 3]) + f2 * v[j + 2];
    }
    __syncthreads();
    // P5: tri-y: bufA (17,17,13) -> bufB (17,13,13): 221 columns
    for (int c = tid; c < 17 * 13; c += 256) {
        int z = c / 13, x = c % 13;
        const float* in = bufA + z * 221 + x;
        float* o = bufB + z * 169 + x;
        float v[17];
#pragma unroll
        for (int j = 0; j < 17; ++j) v[j] = in[j * 13];
#pragma unroll
        for (int j = 0; j < 13; ++j)
            o[j * 13] = f0 * (v[j] + v[j + 4]) + f1 * (v[j + 1] + v[j + 3]) + f2 * v[j + 2];
    }
    __syncthreads();
    // P6: tri-z: bufB (17,13,13) -> global (13,13,13): 169 columns
    for (int c = tid; c < 13 * 13; c += 256) {
        const float* in = bufB + c;
        float v[17];
#pragma unroll
        for (int j = 0; j < 17; ++j) v[j] = in[j * 169];
#pragma unroll
        for (int j = 0; j < 13; ++j)
            outg[j * 169 + c] = f0 * (v[j] + v[j + 4]) + f1 * (v[j + 1] + v[j + 3]) + f2 * v[j + 2];
    }
}

// ---------------------------------------------------------------------------
// WMMA gather: per wave, 16 voxels x 8 neighbors via 2x V_WMMA_F32_16X16X4_F32.
// A (16x4) = broadcast weight rows (ISA layout: lanes0-15 {K0,K1}, lanes16-31 {K2,K3}).
// B (4x16) = gathered neighbor values (mirrored layout).
// D column j (c[0] of lane j) = sum_k w[k] * S[knn[k]][voxbase+j]
// ---------------------------------------------------------------------------
__device__ __forceinline__ float gather16(const float* __restrict__ S,
                                          const int* kn, const float* wn,
                                          int voxbase, int lane) {
    int vox  = voxbase + (lane & 15);
    int voxc = vox < NVOX ? vox : (NVOX - 1);
    bool hi  = lane >= 16;
    v2f a0, a1, b0, b1;
    a0.x = hi ? wn[2] : wn[0];
    a0.y = hi ? wn[3] : wn[1];
    a1.x = hi ? wn[6] : wn[4];
    a1.y = hi ? wn[7] : wn[5];
    b0.x = S[kn[hi ? 2 : 0] * NVOX + voxc];
    b0.y = S[kn[hi ? 3 : 1] * NVOX + voxc];
    b1.x = S[kn[hi ? 6 : 4] * NVOX + voxc];
    b1.y = S[kn[hi ? 7 : 5] * NVOX + voxc];
    v8f c = {};
    c = __builtin_amdgcn_wmma_f32_16x16x4_f32(false, a0, false, b0, (short)0, c, false, false);
    c = __builtin_amdgcn_wmma_f32_16x16x4_f32(false, a1, false, b1, (short)0, c, false, false);
    return c[0];
}

// ---------------------------------------------------------------------------
// K1: S1 = smooth(alpha1 + alpha0 * cost)
// ---------------------------------------------------------------------------
__global__ void __launch_bounds__(256) smooth1_kernel(const float* __restrict__ cost,
                                                      const float* __restrict__ alpha,
                                                      float* __restrict__ S1, int N) {
    __shared__ float cv[NVOX];
    __shared__ float bufA[ASZ];
    __shared__ float bufB[BSZ];
    int n = blockIdx.x, tid = threadIdx.x;
    const float* cn = cost + (size_t)n * NVOX;
    for (int i = tid; i < NVOX; i += 256) async_ld_b32(&cv[i], &cn[i]);
    float a0 = alpha[0], a1 = alpha[1];
    async_wait();
    __syncthreads();
    smooth_lds(cv, bufA, bufB, S1 + (size_t)n * NVOX, tid, a0, a1);
}

// ---------------------------------------------------------------------------
// K2: S2 = smooth(a4 + a2*(a1 + a0*cost) + (a3/norm) * gatherWMMA(S1))
// ---------------------------------------------------------------------------
__global__ void __launch_bounds__(256) gather_smooth_kernel(const float* __restrict__ cost,
                                                            const float* __restrict__ S1,
                                                            const int* __restrict__ knn,
                                                            const float* __restrict__ w,
                                                            const float* __restrict__ nrm,
                                                            const float* __restrict__ alpha,
                                                            float* __restrict__ S2, int N) {
    __shared__ float cv[NVOX];
    __shared__ float bufA[ASZ];
    __shared__ float bufB[BSZ];
    __shared__ int   kn[8];
    __shared__ float wn[8];
    int n = blockIdx.x, tid = threadIdx.x;
    int wave = tid >> 5, lane = tid & 31;
    if (tid < 8) { kn[tid] = knn[n * 8 + tid]; wn[tid] = w[n * 8 + tid]; }
    const float* cn = cost + (size_t)n * NVOX;
    for (int i = tid; i < NVOX; i += 256) async_ld_b32(&cv[i], &cn[i]);
    float a0 = alpha[0], a1 = alpha[1], a2 = alpha[2], a4 = alpha[4];
    float s3 = alpha[3] / nrm[n];
    async_wait();
    __syncthreads();
    // Fused: cv <- a4 + a2*(a1 + a0*cv) + s3 * (WMMA KNN gather of S1)
    for (int t = wave; t < NTILE; t += 8) {
        int base = t * 16;
        float g = gather16(S1, kn, wn, base, lane);
        int vox = base + (lane & 15);
        if (lane < 16 && vox < NVOX)
            cv[vox] = a4 + a2 * (a1 + a0 * cv[vox]) + s3 * g;
    }
    __syncthreads();
    smooth_lds(cv, bufA, bufB, S2 + (size_t)n * NVOX, tid, 1.0f, 0.0f);
}

// ---------------------------------------------------------------------------
// K3: out = (alpha5 / norm) * gatherWMMA(S2)   (nontemporal final stores)
// ---------------------------------------------------------------------------
__global__ void __launch_bounds__(256) gather_out_kernel(const float* __restrict__ S2,
                                                         const int* __restrict__ knn,
                                                         const float* __restrict__ w,
                                                         const float* __restrict__ nrm,
                                                         const float* __restrict__ alpha,
                                                         float* __restrict__ out, int N) {
    __shared__ int   kn[8];
    __shared__ float wn[8];
    __shared__ float sc;
    int n = blockIdx.x, tid = threadIdx.x;
    int wave = tid >> 5, lane = tid & 31;
    if (tid < 8) { kn[tid] = knn[n * 8 + tid]; wn[tid] = w[n * 8 + tid]; }
    if (tid == 8) sc = alpha[5] / nrm[n];
    __syncthreads();
    float scale = sc;
    float* on = out + (size_t)n * NVOX;
    for (int t = wave; t < NTILE; t += 8) {
        int base = t * 16;
        float v = gather16(S2, kn, wn, base, lane);
        int vox = base + (lane & 15);
        if (lane < 16 && vox < NVOX)
            __builtin_nontemporal_store(scale * v, &on[vox]);
    }
}

// ---------------------------------------------------------------------------
extern "C" void kernel_launch(void* const* d_in, const int* in_sizes, int n_in,
                              void* d_out, int out_size, void* d_ws, size_t ws_size,
                              hipStream_t stream) {
    const float* cost  = (const float*)d_in[0];
    const int*   knn   = (const int*)d_in[1];
    const float* dist  = (const float*)d_in[2];
    const float* alpha = (const float*)d_in[3];
    int N = in_sizes[0] / NVOX;           // 10000

    float* S1  = (float*)d_ws;
    float* S2  = S1 + (size_t)N * NVOX;
    float* w   = S2 + (size_t)N * NVOX;
    float* nrm = w + (size_t)N * 8;

    wnorm_kernel<<<(N + 255) / 256, 256, 0, stream>>>(dist, w, nrm, N);
    smooth1_kernel<<<N, 256, 0, stream>>>(cost, alpha, S1, N);
    gather_smooth_kernel<<<N, 256, 0, stream>>>(cost, S1, knn, w, nrm, alpha, S2, N);
    gather_out_kernel<<<N, 256, 0, stream>>>(S2, knn, w, nrm, alpha, (float*)d_out, N);
}